// myDC_14396730376405
// MI455X (gfx1250) — compile-verified
//
#include <hip/hip_runtime.h>

// ---------------- problem dimensions ----------------
#define NRr 200
#define NCc 200
#define NSs 8
#define COILS 8
#define NRP 240
#define HZ 20
#define YX 1600L            // NCc*NSs
#define VV 320000L          // NRr*NCc*NSs
#define PP 384000L          // NRP*NCc*NSs
#define CPP 3072000L        // COILS*PP
#define CG_TOL 1e-5f
#define CG_ITERS 10

typedef float v2f __attribute__((ext_vector_type(2)));
typedef float v8f __attribute__((ext_vector_type(8)));
typedef unsigned int u32x4 __attribute__((ext_vector_type(4)));
typedef int i32x4 __attribute__((ext_vector_type(4)));
typedef int i32x8 __attribute__((ext_vector_type(8)));

// ---------------- DFT matrix fill ----------------
__global__ void fill_dft(float* __restrict__ wr, float* __restrict__ wi,
                         int N, float sign, float scale) {
    long i = (long)blockIdx.x * blockDim.x + threadIdx.x;
    if (i >= (long)N * N) return;
    int m = (int)(i / N), k = (int)(i % N);
    double ang = (double)sign * 6.283185307179586 * (double)(((long)m * k) % N) / (double)N;
    wr[i] = (float)cos(ang) * scale;
    wi[i] = (float)sin(ang) * scale;
}

// ---------------- TDM panel stage: 16 x rowLen f32 tile, row stride rowLen ----
// tensor_dim1 = rowsAvail (< 16 on tail tiles) => HW returns zeros for OOB rows.
__device__ inline void tdm_load_panel(const float* gsrc, unsigned ldsByteOff,
                                      int rowLen, int rowsAvail) {
    unsigned long long ga = (unsigned long long)(const void*)gsrc;
    u32x4 g0;
    g0[0] = 1u;                                              // count=1, user mode
    g0[1] = ldsByteOff;                                      // lds_addr (bytes)
    g0[2] = (unsigned)(ga & 0xffffffffu);                    // global_addr[31:0]
    g0[3] = (unsigned)((ga >> 32) & 0x01ffffffu) | (2u << 30); // addr[56:32] | type=2
    i32x8 g1;
    g1[0] = (int)(2u << 16);                                 // data_size=4B, mask=0
    g1[1] = (int)((unsigned)(rowLen & 0xffff) << 16);        // tensor_dim0[15:0]
    g1[2] = (int)(((unsigned)rowLen >> 16) |
                  ((unsigned)(rowsAvail & 0xffff) << 16));   // dim0 hi | tensor_dim1 lo
    g1[3] = (int)(((unsigned)rowsAvail >> 16) |
                  ((unsigned)(rowLen & 0xffff) << 16));      // dim1 hi | tile_dim0
    g1[4] = 16;                                              // tile_dim1=16, tile_dim2=0
    g1[5] = rowLen;                                          // tensor_dim0_stride lo
    g1[6] = 0;                                               // stride hi | dim1_stride lo
    g1[7] = 0;                                               // dim1_stride hi
    i32x4 z4 = {0, 0, 0, 0};
#if defined(__clang_major__) && (__clang_major__ >= 23)
    i32x8 z8 = {0, 0, 0, 0, 0, 0, 0, 0};
    __builtin_amdgcn_tensor_load_to_lds(g0, g1, z4, z4, z8, 0);
#else
    __builtin_amdgcn_tensor_load_to_lds(g0, g1, z4, z4, 0);
#endif
}

// ---------------- WMMA complex DFT pass ----------------
// out[m][n] = sum_k W[m][k] * in[k][n]   (complex, split planes)
// element address for (k or m, n): (n/innerN)*outerStride + (n%innerN) + k*sk
// Host guarantees: Ndft % 4 == 0, nTotal % 128 == 0 (no n guards needed).
// One wave computes a 16(m) x 32(n) output tile = 2 WMMA tiles sharing A.
__global__ __launch_bounds__(128)
void cdft_wmma(float* __restrict__ oRe, float* __restrict__ oIm,
               const float* __restrict__ iRe, const float* __restrict__ iIm,
               const float* __restrict__ wRe, const float* __restrict__ wIm,
               int Ndft, long nTotal, int innerN, long outerStride, int sk)
{
    extern __shared__ float smem[];                 // [16*Ndft] Wre | [16*Ndft] Wim
    const int m0    = blockIdx.y * 16;
    const int total = 16 * Ndft;

    // Stage the 16-row W panel into LDS with the Tensor Data Mover (wave 0 only).
    if (threadIdx.x < 32) {
        int rowsAvail = Ndft - m0;                  // tail tiles rely on TDM OOB->0
        tdm_load_panel(wRe + (long)m0 * Ndft, 0u, Ndft, rowsAvail);
        tdm_load_panel(wIm + (long)m0 * Ndft, (unsigned)(total * 4), Ndft, rowsAvail);
        __builtin_amdgcn_s_wait_tensorcnt(0);
    }
    __syncthreads();

    const int wid  = threadIdx.x >> 5;
    const int lane = threadIdx.x & 31;
    const long n0  = ((long)blockIdx.x * 4 + wid) * 32;

    const int  mIdx = lane & 15;
    const int  kSel = (lane >> 4) << 1;            // 0 or 2 (ISA 16x4 f32 A layout)
    const long nIdxA = n0 + (lane & 15);
    const long nIdxB = nIdxA + 16;
    const long baseA = (nIdxA / innerN) * outerStride + (nIdxA % innerN);
    const long baseB = (nIdxB / innerN) * outerStride + (nIdxB % innerN);
    const float* lre = smem + (long)mIdx * Ndft;
    const float* lim = smem + total + (long)mIdx * Ndft;

    v8f accRe0 = {0.f,0.f,0.f,0.f,0.f,0.f,0.f,0.f};
    v8f accIm0 = {0.f,0.f,0.f,0.f,0.f,0.f,0.f,0.f};
    v8f accRe1 = {0.f,0.f,0.f,0.f,0.f,0.f,0.f,0.f};
    v8f accIm1 = {0.f,0.f,0.f,0.f,0.f,0.f,0.f,0.f};

    for (int k0 = 0; k0 < Ndft; k0 += 4) {
        const int ka = k0 + kSel, kb = ka + 1;
        v2f aRe = { lre[ka], lre[kb] };
        v2f aIm = { lim[ka], lim[kb] };
        v2f aImN = { -aIm.x, -aIm.y };
        const long oa0 = baseA + (long)ka * sk, oa1 = baseA + (long)kb * sk;
        const long ob0 = baseB + (long)ka * sk, ob1 = baseB + (long)kb * sk;
        v2f bRe0 = { iRe[oa0], iRe[oa1] };
        v2f bIm0 = { iIm[oa0], iIm[oa1] };
        v2f bRe1 = { iRe[ob0], iRe[ob1] };
        v2f bIm1 = { iIm[ob0], iIm[ob1] };
        // complex MAC = 4 real f32 WMMAs per n-tile; A fragments reused across tiles
        accRe0 = __builtin_amdgcn_wmma_f32_16x16x4_f32(false, aRe,  false, bRe0, (short)0, accRe0, false, false);
        accRe0 = __builtin_amdgcn_wmma_f32_16x16x4_f32(false, aImN, false, bIm0, (short)0, accRe0, false, false);
        accIm0 = __builtin_amdgcn_wmma_f32_16x16x4_f32(false, aRe,  false, bIm0, (short)0, accIm0, false, false);
        accIm0 = __builtin_amdgcn_wmma_f32_16x16x4_f32(false, aIm,  false, bRe0, (short)0, accIm0, false, false);
        accRe1 = __builtin_amdgcn_wmma_f32_16x16x4_f32(false, aRe,  false, bRe1, (short)0, accRe1, false, false);
        accRe1 = __builtin_amdgcn_wmma_f32_16x16x4_f32(false, aImN, false, bIm1, (short)0, accRe1, false, false);
        accIm1 = __builtin_amdgcn_wmma_f32_16x16x4_f32(false, aRe,  false, bIm1, (short)0, accIm1, false, false);
        accIm1 = __builtin_amdgcn_wmma_f32_16x16x4_f32(false, aIm,  false, bRe1, (short)0, accIm1, false, false);
    }

    const int mRowOff = (lane < 16) ? 0 : 8;       // ISA 16x16 f32 C/D layout
    #pragma unroll
    for (int v = 0; v < 8; ++v) {
        int m = m0 + v + mRowOff;
        if (m < Ndft) {
            long oA = baseA + (long)m * sk;
            long oB = baseB + (long)m * sk;
            oRe[oA] = accRe0[v];
            oIm[oA] = accIm0[v];
            oRe[oB] = accRe1[v];
            oIm[oB] = accIm1[v];
        }
    }
}

// ---------------- pointwise kernels ----------------
__global__ void expand_pad(float* __restrict__ br, float* __restrict__ bi,
                           const float* __restrict__ csm,
                           const float* __restrict__ pr, const float* __restrict__ pi) {
    long i = (long)blockIdx.x * blockDim.x + threadIdx.x;
    if (i >= CPP) return;
    int  c   = (int)(i / PP);
    long p   = i % PP;
    int  z   = (int)(p / YX);
    int  rem = (int)(p % YX);
    float re = 0.f, im = 0.f;
    int zr = z - HZ;
    if (zr >= 0 && zr < NRr) {
        long vi = (long)zr * YX + rem;
        long ci = (((long)c * NRr + zr) * YX + rem) * 2;
        float cr = csm[ci], cim = csm[ci + 1];
        float xr = pr[vi], xi = pi[vi];
        re = cr * xr - cim * xi;
        im = cr * xi + cim * xr;
    }
    br[i] = re; bi[i] = im;
}

__global__ void wpsf_mult(float* __restrict__ dr, float* __restrict__ di,
                          const float* __restrict__ sr, const float* __restrict__ si,
                          const float* __restrict__ wpsf, float sgn) {
    long i = (long)blockIdx.x * blockDim.x + threadIdx.x;
    if (i >= CPP) return;
    long p = i % PP;
    float wr = wpsf[p * 2], wi = sgn * wpsf[p * 2 + 1];
    float xr = sr[i], xi = si[i];
    dr[i] = wr * xr - wi * xi;
    di[i] = wr * xi + wi * xr;
}

__global__ void mask_mult(float* __restrict__ dr, float* __restrict__ di,
                          const float* __restrict__ sr, const float* __restrict__ si,
                          const int* __restrict__ mask) {
    long i = (long)blockIdx.x * blockDim.x + threadIdx.x;
    if (i >= CPP) return;
    float m = (float)mask[i % PP];
    dr[i] = sr[i] * m;
    di[i] = si[i] * m;
}

__global__ void combine_crop(float* __restrict__ apr, float* __restrict__ api,
                             const float* __restrict__ br, const float* __restrict__ bi,
                             const float* __restrict__ csm,
                             const float* __restrict__ pr, const float* __restrict__ pi,
                             const float* __restrict__ lam1, const float* __restrict__ lam2) {
    long i = (long)blockIdx.x * blockDim.x + threadIdx.x;
    if (i >= VV) return;
    int zr  = (int)(i / YX);
    int rem = (int)(i % YX);
    float sre = 0.f, sim = 0.f;
    #pragma unroll
    for (int c = 0; c < COILS; ++c) {
        long bidx = ((long)c * NRP + (zr + HZ)) * YX + rem;
        float vr = br[bidx], vi = bi[bidx];
        long ci = (((long)c * NRr + zr) * YX + rem) * 2;
        float cr = csm[ci], cim = csm[ci + 1];
        sre += cr * vr + cim * vi;     // conj(csm)*v
        sim += cr * vi - cim * vr;
    }
    float l3 = lam1[0] + lam2[0];
    apr[i] = sre + l3 * pr[i];
    api[i] = sim + l3 * pi[i];
}

// ---------------- CG support ----------------
__global__ void cg_init(float* xr, float* xi, float* rr, float* ri,
                        float* pr, float* pi, const float* __restrict__ rhs) {
    long i = (long)blockIdx.x * blockDim.x + threadIdx.x;
    if (i >= VV) return;
    float re = rhs[2 * i], im = rhs[2 * i + 1];
    xr[i] = 0.f; xi[i] = 0.f;
    rr[i] = re;  ri[i] = im;
    pr[i] = re;  pi[i] = im;
}

__global__ void zero1(float* p) { *p = 0.f; }

__global__ void dot_re(const float* __restrict__ ar, const float* __restrict__ ai,
                       const float* __restrict__ br, const float* __restrict__ bi,
                       float* __restrict__ out, long n) {
    __shared__ float red[256];
    long stride = (long)gridDim.x * blockDim.x;
    float v = 0.f;
    for (long j = (long)blockIdx.x * blockDim.x + threadIdx.x; j < n; j += stride)
        v += ar[j] * br[j] + ai[j] * bi[j];       // Re(conj(a)*b)
    red[threadIdx.x] = v;
    __syncthreads();
    for (int s = 128; s > 0; s >>= 1) {
        if ((int)threadIdx.x < s) red[threadIdx.x] += red[threadIdx.x + s];
        __syncthreads();
    }
    if (threadIdx.x == 0) atomicAdd(out, red[0]);
}

__global__ void update_xr(float* xr, float* xi, float* rr, float* ri,
                          const float* pr, const float* pi,
                          const float* apr, const float* api, const float* s) {
    long i = (long)blockIdx.x * blockDim.x + threadIdx.x;
    if (i >= VV) return;
    float rtr = s[0], pap = s[1];
    float a = (rtr > CG_TOL) ? (rtr / pap) : 0.f;
    xr[i] += a * pr[i];  xi[i] += a * pi[i];
    rr[i] -= a * apr[i]; ri[i] -= a * api[i];
}

__global__ void update_p(float* pr, float* pi,
                         const float* rr, const float* ri, const float* s) {
    long i = (long)blockIdx.x * blockDim.x + threadIdx.x;
    if (i >= VV) return;
    float rtr = s[0];
    if (rtr > CG_TOL) {
        float b = s[2] / rtr;
        pr[i] = rr[i] + b * pr[i];
        pi[i] = ri[i] + b * pi[i];
    }
}

__global__ void advance_s(float* s) { if (s[0] > CG_TOL) s[0] = s[2]; }

__global__ void writeout(float* __restrict__ out,
                         const float* __restrict__ xr, const float* __restrict__ xi) {
    long i = (long)blockIdx.x * blockDim.x + threadIdx.x;
    if (i >= VV) return;
    out[2 * i]     = xr[i];
    out[2 * i + 1] = xi[i];
}

// ---------------- host orchestration ----------------
extern "C" void kernel_launch(void* const* d_in, const int* in_sizes, int n_in,
                              void* d_out, int out_size, void* d_ws, size_t ws_size,
                              hipStream_t stream) {
    (void)in_sizes; (void)n_in; (void)out_size; (void)ws_size;
    const float* rhs  = (const float*)d_in[0];
    const float* csm  = (const float*)d_in[1];
    const float* wpsf = (const float*)d_in[2];
    const float* lam1 = (const float*)d_in[3];
    const float* lam2 = (const float*)d_in[4];
    const int*   mask = (const int*)  d_in[5];
    float* out = (float*)d_out;

    float* ws = (float*)d_ws;
    float* xr  = ws + 0 * VV; float* xi  = ws + 1 * VV;
    float* rr  = ws + 2 * VV; float* ri  = ws + 3 * VV;
    float* pr  = ws + 4 * VV; float* pi  = ws + 5 * VV;
    float* apr = ws + 6 * VV; float* api = ws + 7 * VV;
    float* bAr = ws + 8 * VV;  float* bAi = bAr + CPP;
    float* bBr = bAi + CPP;    float* bBi = bBr + CPP;
    float* w240fr = bBi + CPP;      float* w240fi = w240fr + 57600;
    float* w240ir = w240fi + 57600; float* w240ii = w240ir + 57600;
    float* w200fr = w240ii + 57600; float* w200fi = w200fr + 40000;
    float* w200ir = w200fi + 40000; float* w200ii = w200ir + 40000;
    float* w8fr = w200ii + 40000;   float* w8fi = w8fr + 64;
    float* w8ir = w8fi + 64;        float* w8ii = w8ir + 64;
    float* s = w8ii + 64;

    const int T = 256;
    const long gV = (VV + T - 1) / T, gCP = (CPP + T - 1) / T;

    // DFT matrices (forward e^{-i}, inverse e^{+i}/N)
    fill_dft<<<(57600 + T - 1) / T, T, 0, stream>>>(w240fr, w240fi, 240, -1.f, 1.f);
    fill_dft<<<(57600 + T - 1) / T, T, 0, stream>>>(w240ir, w240ii, 240,  1.f, 1.f / 240.f);
    fill_dft<<<(40000 + T - 1) / T, T, 0, stream>>>(w200fr, w200fi, 200, -1.f, 1.f);
    fill_dft<<<(40000 + T - 1) / T, T, 0, stream>>>(w200ir, w200ii, 200,  1.f, 1.f / 200.f);
    fill_dft<<<1, 64, 0, stream>>>(w8fr, w8fi, 8, -1.f, 1.f);
    fill_dft<<<1, 64, 0, stream>>>(w8ir, w8ii, 8,  1.f, 0.125f);

    auto dft = [&](float* oR, float* oI, const float* iR, const float* iI,
                   const float* wR, const float* wI,
                   int Nd, long nTot, int inn, long outS, int sk) {
        // 4 waves/block, 32 columns per wave => 128 columns per block
        dim3 grid((unsigned)((nTot + 127) / 128), (unsigned)((Nd + 15) / 16));
        size_t sh = (size_t)2 * 16 * Nd * sizeof(float);
        cdft_wmma<<<grid, dim3(128), sh, stream>>>(oR, oI, iR, iI, wR, wI,
                                                   Nd, nTot, inn, outS, sk);
    };

    // CG init
    cg_init<<<gV, T, 0, stream>>>(xr, xi, rr, ri, pr, pi, rhs);
    zero1<<<1, 1, 0, stream>>>(s + 0);
    dot_re<<<512, T, 0, stream>>>(rr, ri, rr, ri, s + 0, VV);

    for (int it = 0; it < CG_ITERS; ++it) {
        // ---- Ap = A(p):  FFT_z -> xwPSF -> FFT_y FFT_x -> xmask
        //                  -> IFFT_x IFFT_y -> xconj(wPSF) -> IFFT_z -> crop/combine
        expand_pad<<<gCP, T, 0, stream>>>(bAr, bAi, csm, pr, pi);
        dft(bBr, bBi, bAr, bAi, w240fr, w240fi, 240, 12800L, 1600, PP, 1600);   // z fwd
        wpsf_mult<<<gCP, T, 0, stream>>>(bAr, bAi, bBr, bBi, wpsf, 1.f);
        dft(bBr, bBi, bAr, bAi, w200fr, w200fi, 200, 15360L, 8, 1600L, 8);      // y fwd
        dft(bAr, bAi, bBr, bBi, w8fr, w8fi, 8, 384000L, 1, 8L, 1);              // x fwd
        mask_mult<<<gCP, T, 0, stream>>>(bBr, bBi, bAr, bAi, mask);
        dft(bAr, bAi, bBr, bBi, w8ir, w8ii, 8, 384000L, 1, 8L, 1);              // x inv
        dft(bBr, bBi, bAr, bAi, w200ir, w200ii, 200, 15360L, 8, 1600L, 8);      // y inv
        wpsf_mult<<<gCP, T, 0, stream>>>(bAr, bAi, bBr, bBi, wpsf, -1.f);
        dft(bBr, bBi, bAr, bAi, w240ir, w240ii, 240, 12800L, 1600, PP, 1600);   // z inv
        combine_crop<<<gV, T, 0, stream>>>(apr, api, bBr, bBi, csm, pr, pi, lam1, lam2);

        // ---- CG scalar updates (all on-device; graph-capture safe)
        zero1<<<1, 1, 0, stream>>>(s + 1);
        dot_re<<<512, T, 0, stream>>>(pr, pi, apr, api, s + 1, VV);
        update_xr<<<gV, T, 0, stream>>>(xr, xi, rr, ri, pr, pi, apr, api, s);
        zero1<<<1, 1, 0, stream>>>(s + 2);
        dot_re<<<512, T, 0, stream>>>(rr, ri, rr, ri, s + 2, VV);
        update_p<<<gV, T, 0, stream>>>(pr, pi, rr, ri, s);
        advance_s<<<1, 1, 0, stream>>>(s);
    }

    writeout<<<gV, T, 0, stream>>>(out, xr, xi);
}